// STRNN_24885040513384
// MI455X (gfx1250) — compile-verified
//
#include <hip/hip_runtime.h>
#include <hip/hip_bf16.h>

// STRNN collapsed-sum implementation for MI455X (gfx1250, wave32).
// All heavy matmuls collapse through sum(s); kernel chain is pure bandwidth
// (~115 MB -> ~5us roofline at 23.3 TB/s). CDNA5 paths used:
// global_load_async_to_lds_b128 + s_wait_asynccnt (inline asm, portable
// across toolchains) and non-temporal streaming loads for stream-once data.

#define HH 2048   // hidden size
#define LL 4096   // trajectory length
#define NB3 256   // rowdot blocks (== threads/block for final reduce)

// Native Clang vector type: required by __builtin_nontemporal_load
// (HIP's float4 is a struct and is rejected).
typedef float vf4 __attribute__((ext_vector_type(4)));

// ---------------------------------------------------------------------------
// Deterministic block reduction (256 threads, wave32). Result valid in t==0.
__device__ __forceinline__ float block_reduce_256(float v, float* sm) {
#pragma unroll
  for (int off = 16; off > 0; off >>= 1) v += __shfl_down(v, off, 32);
  const int lane = threadIdx.x & 31;
  const int wid  = threadIdx.x >> 5;
  if (lane == 0) sm[wid] = v;
  __syncthreads();
  float r = 0.f;
  if (wid == 0) {
    r = (lane < 8) ? sm[lane] : 0.f;
#pragma unroll
    for (int off = 4; off > 0; off >>= 1) r += __shfl_down(r, off, 32);
  }
  __syncthreads();   // allow safe reuse of sm
  return r;
}

// ---------------------------------------------------------------------------
// 1) Partial column sums of W_sh_upper / W_sh_lower.
//    csp[(m*JS+z)*HH + k] = sum over row-chunk z of W[m][j][k]
template <int JS>
__global__ void strnn_cs(const float* __restrict__ Wsu,
                         const float* __restrict__ Wsl,
                         float* __restrict__ csp) {
  constexpr int CH = HH / JS;
  const int k = blockIdx.x * 256 + threadIdx.x;
  const int m = blockIdx.y;
  const int z = blockIdx.z;
  const float* __restrict__ M = (m ? Wsl : Wsu) + (size_t)z * CH * HH + k;
  float s = 0.f;
#pragma unroll 8
  for (int j = 0; j < CH; ++j)
    s += __builtin_nontemporal_load(M + (size_t)j * HH);
  csp[((size_t)m * JS + z) * HH + k] = s;
}

// ---------------------------------------------------------------------------
// 2) Partial p-vectors: pvp[((m*JS+z)*2+o)*HH + k]
//    m=0 -> W_th_upper, m=1 -> W_th_lower; o=0 weighted by w_up, o=1 by w_lo.
template <int JS>
__global__ void strnn_pv(const float* __restrict__ Wtu,
                         const float* __restrict__ Wtl,
                         const float* __restrict__ csp,
                         float* __restrict__ pvp) {
  constexpr int CH = HH / JS;
  __shared__ float wu_s[CH];
  __shared__ float wl_s[CH];
  const int t = threadIdx.x;
  const int k = blockIdx.x * 256 + t;
  const int m = blockIdx.y;
  const int z = blockIdx.z;
  const int j0 = z * CH;

  // Finalize the needed chunk of the column-sum vectors (reads tiny csp).
  for (int jj = t; jj < CH; jj += 256) {
    float su = 0.f, sl = 0.f;
#pragma unroll
    for (int zz = 0; zz < JS; ++zz) {
      su += csp[(size_t)zz * HH + j0 + jj];
      sl += csp[(size_t)(JS + zz) * HH + j0 + jj];
    }
    wu_s[jj] = su;
    wl_s[jj] = sl;
  }
  __syncthreads();

  const float* __restrict__ M = (m ? Wtl : Wtu) + (size_t)j0 * HH + k;
  float s0 = 0.f, s1 = 0.f;
#pragma unroll 8
  for (int jj = 0; jj < CH; ++jj) {
    const float v = __builtin_nontemporal_load(M + (size_t)jj * HH);
    s0 += v * wu_s[jj];
    s1 += v * wl_s[jj];
  }
  float* __restrict__ o = pvp + ((size_t)m * JS + z) * 2 * HH + k;
  o[0]  = s0;
  o[HH] = s1;
}

// ---------------------------------------------------------------------------
// 3) Reduce partial p-vectors -> P[4][HH]  (P0=p_uu, P1=p_ul, P2=p_lu, P3=p_ll)
template <int JS>
__global__ void strnn_pvr(const float* __restrict__ pvp, float* __restrict__ P) {
  const int idx = blockIdx.x * 256 + threadIdx.x;  // 0 .. 4*HH-1
  const int vec = idx >> 11;                       // /HH
  const int k   = idx & (HH - 1);
  const int m = vec >> 1, o = vec & 1;
  float s = 0.f;
#pragma unroll
  for (int z = 0; z < JS; ++z)
    s += pvp[(((size_t)m * JS + z) * 2 + o) * HH + k];
  P[idx] = s;
}

// ---------------------------------------------------------------------------
// 4) Gather rows of location_weight and accumulate weighted 4-way dots.
//    P (32 KB) is staged into LDS with CDNA5 async global->LDS b128 copies.
__global__ void strnn_rowdot(const float* __restrict__ lw,
                             const int* __restrict__ loc,
                             const float* __restrict__ tdu,
                             const float* __restrict__ tdl,
                             const float* __restrict__ ldu,
                             const float* __restrict__ ldl,
                             const float* __restrict__ P,
                             float* __restrict__ partials) {
  __shared__ __align__(16) float Pq[4 * HH];  // 32 KB
  __shared__ float red[8];
  const int t = threadIdx.x;

  // Async copy: 8 x b128 per thread (256 threads x 8 x 16B = 32 KB).
#pragma unroll
  for (int j = 0; j < (4 * HH) / (256 * 4); ++j) {
    const int f = (t + j * 256) * 4;
#if defined(__AMDGCN__)
    unsigned lds_off = (unsigned)(size_t)(&Pq[f]);  // flat LDS addr low 32 bits
    const float* g = P + f;
    asm volatile("global_load_async_to_lds_b128 %0, %1, off"
                 :: "v"(lds_off), "v"(g) : "memory");
#else
    Pq[f]     = P[f];
    Pq[f + 1] = P[f + 1];
    Pq[f + 2] = P[f + 2];
    Pq[f + 3] = P[f + 3];
#endif
  }
#if defined(__AMDGCN__)
  asm volatile("s_wait_asynccnt 0" ::: "memory");
#endif
  __syncthreads();

  float acc = 0.f;
  for (int r = blockIdx.x; r < LL; r += NB3) {
    const float tu = tdu[r], tl = tdl[r], lu = ldu[r], llw = ldl[r];
    const float it = 1.f / (tu + tl);
    const float il = 1.f / (lu + llw);
    const float a = tu * it, b = tl * it, c = lu * il, d = llw * il;
    const float ac = a * c, ad = a * d, bc = b * c, bd = b * d;
    const float* __restrict__ row = lw + (size_t)loc[r] * HH;
#pragma unroll
    for (int j = 0; j < 2; ++j) {
      const int k = (t + j * 256) * 4;
      const vf4 v  = __builtin_nontemporal_load((const vf4*)(row + k));
      const vf4 p0 = *(const vf4*)(&Pq[k]);
      const vf4 p1 = *(const vf4*)(&Pq[HH + k]);
      const vf4 p2 = *(const vf4*)(&Pq[2 * HH + k]);
      const vf4 p3 = *(const vf4*)(&Pq[3 * HH + k]);
      acc += v.x * (ac * p0.x + ad * p1.x + bc * p2.x + bd * p3.x);
      acc += v.y * (ac * p0.y + ad * p1.y + bc * p2.y + bd * p3.y);
      acc += v.z * (ac * p0.z + ad * p1.z + bc * p2.z + bd * p3.z);
      acc += v.w * (ac * p0.w + ad * p1.w + bc * p2.w + bd * p3.w);
    }
  }
  const float bs = block_reduce_256(acc, red);
  if (t == 0) partials[blockIdx.x] = bs;
}

// ---------------------------------------------------------------------------
// 5) out[row] = sigmoid(loc_vec + W_ih[row,:] . hx). Each block re-reduces the
//    256 partials in an identical fixed order -> bit-identical loc_vec.
__global__ void strnn_final(const float* __restrict__ Wih,
                            const float* __restrict__ hx,
                            const float* __restrict__ partials,
                            float* __restrict__ out) {
  __shared__ float red[8];
  const int t = threadIdx.x;
  const int row = blockIdx.x;

  const float lv = block_reduce_256(partials[t], red);  // valid in t==0

  const float* __restrict__ wr = Wih + (size_t)row * HH;
  float acc = 0.f;
#pragma unroll
  for (int j = 0; j < 2; ++j) {
    const int k = (t + j * 256) * 4;
    const vf4 w = __builtin_nontemporal_load((const vf4*)(wr + k));
    const vf4 h = *(const vf4*)(hx + k);
    acc += w.x * h.x + w.y * h.y + w.z * h.z + w.w * h.w;
  }
  const float dv = block_reduce_256(acc, red);
  if (t == 0) {
    const float x = lv + dv;
    out[row] = 1.f / (1.f + expf(-x));
  }
}

// ---------------------------------------------------------------------------
template <int JS>
static void strnn_run(const float* tdu, const float* tdl,
                      const float* ldu, const float* ldl,
                      const float* hx, const float* Wih,
                      const float* Wtu, const float* Wtl,
                      const float* Wsu, const float* Wsl,
                      const float* lw, const int* loc,
                      float* ws, float* out, hipStream_t stream) {
  float* csp      = ws;                                  // 2*JS*HH
  float* pvp      = csp + (size_t)2 * JS * HH;           // 4*JS*HH
  float* P        = pvp + (size_t)4 * JS * HH;           // 4*HH
  float* partials = P + 4 * HH;                          // NB3

  strnn_cs<JS><<<dim3(HH / 256, 2, JS), 256, 0, stream>>>(Wsu, Wsl, csp);
  strnn_pv<JS><<<dim3(HH / 256, 2, JS), 256, 0, stream>>>(Wtu, Wtl, csp, pvp);
  strnn_pvr<JS><<<dim3((4 * HH) / 256), 256, 0, stream>>>(pvp, P);
  strnn_rowdot<<<dim3(NB3), 256, 0, stream>>>(lw, loc, tdu, tdl, ldu, ldl, P,
                                              partials);
  strnn_final<<<dim3(HH), 256, 0, stream>>>(Wih, hx, partials, out);
}

extern "C" void kernel_launch(void* const* d_in, const int* in_sizes, int n_in,
                              void* d_out, int out_size, void* d_ws,
                              size_t ws_size, hipStream_t stream) {
  const float* tdu = (const float*)d_in[0];
  const float* tdl = (const float*)d_in[1];
  const float* ldu = (const float*)d_in[2];
  const float* ldl = (const float*)d_in[3];
  const float* hx  = (const float*)d_in[4];
  const float* Wih = (const float*)d_in[5];
  const float* Wtu = (const float*)d_in[6];
  const float* Wtl = (const float*)d_in[7];
  const float* Wsu = (const float*)d_in[8];
  const float* Wsl = (const float*)d_in[9];
  const float* lw  = (const float*)d_in[10];
  const int*   loc = (const int*)d_in[11];
  float* out = (float*)d_out;
  float* ws  = (float*)d_ws;

  auto need = [](size_t js) {
    return sizeof(float) * (6 * js * (size_t)HH + 4 * (size_t)HH + NB3);
  };
  if (ws_size >= need(16)) {
    strnn_run<16>(tdu, tdl, ldu, ldl, hx, Wih, Wtu, Wtl, Wsu, Wsl, lw, loc, ws,
                  out, stream);
  } else if (ws_size >= need(8)) {
    strnn_run<8>(tdu, tdl, ldu, ldl, hx, Wih, Wtu, Wtl, Wsu, Wsl, lw, loc, ws,
                 out, stream);
  } else {
    strnn_run<1>(tdu, tdl, ldu, ldl, hx, Wih, Wtu, Wtl, Wsu, Wsl, lw, loc, ws,
                 out, stream);
  }
}